// AgnNetwork_83726092468412
// MI455X (gfx1250) — compile-verified
//
#include <hip/hip_runtime.h>
#include <hip/hip_bf16.h>
#include <math.h>

typedef __attribute__((ext_vector_type(2))) float v2f;
typedef __attribute__((ext_vector_type(8))) float v8f;

#define HID 64
#define NCLS 40
#define FIN 128
#define EPS 1e-12f

// ---------------- utility: wave32 reductions ----------------
__device__ __forceinline__ float waveReduceSum(float v) {
#pragma unroll
    for (int off = 16; off > 0; off >>= 1) v += __shfl_xor(v, off, 32);
    return v;
}
__device__ __forceinline__ float waveReduceMax(float v) {
#pragma unroll
    for (int off = 16; off > 0; off >>= 1) v = fmaxf(v, __shfl_xor(v, off, 32));
    return v;
}

// float atomic max via signed/unsigned int trick (works with -inf init),
// avoids dependence on HIP header float atomicMax availability.
__device__ __forceinline__ void atomicMaxF(float* addr, float value) {
    if (value >= 0.0f)
        atomicMax((int*)addr, __float_as_int(value));
    else
        atomicMin((unsigned int*)addr, __float_as_uint(value));
}

// ---------------- WMMA f32 GEMM: C[M x Ncols] = A[M x K] @ B[K x Ncols] + bias (+relu) ----------
// One wave computes one 16x16 tile via V_WMMA_F32_16X16X4_F32 along K.
// Layouts per CDNA5 ISA 7.12.2:
//   A 16x4 : lane(0-15) M=lane, VGPR0/1 = K0/K1 ; lane(16-31) M=lane-16, VGPR0/1 = K2/K3
//   B 4x16 : VGPRv lanes0-15 = row K=v, N=lane ; lanes16-31 = row K=v+2, N=lane-16
//   C 16x16: VGPRv lanes0-15 = M=v, N=lane    ; lanes16-31 = M=v+8, N=lane-16
// Branch-free inner loop: out-of-range N columns are clamped for the address and
// zeroed via a multiplicative mask, so no per-iteration EXEC manipulation.
__global__ void gemm_wmma_f32(const float* __restrict__ A, const float* __restrict__ B,
                              const float* __restrict__ bias, float* __restrict__ C,
                              int M, int K, int Ncols, int n_tiles, int do_relu) {
    const int wave = (blockIdx.x * blockDim.x + threadIdx.x) >> 5;
    const int lane = threadIdx.x & 31;
    const int total = (M >> 4) * n_tiles;
    if (wave >= total) return;   // wave-uniform: EXEC all-1s inside
    const int mt = wave / n_tiles;
    const int nt = wave - mt * n_tiles;
    const int m0 = mt << 4;
    const int n0 = nt << 4;

    const int lrow  = lane & 15;
    const int khalf = (lane >> 4) << 1;         // 0 or 2

    const float* arow = A + (size_t)(m0 + lrow) * K;    // A row for this lane
    const int bn = n0 + lrow;                           // B/C column for this lane
    const bool nvalid = (bn < Ncols);
    const int bnc = nvalid ? bn : (Ncols - 1);          // clamped (safe) column
    const float bmask = nvalid ? 1.0f : 0.0f;           // zero out padded columns
    const float* bcol = B + bnc;

    v8f acc = {};
#pragma unroll 2
    for (int kb = 0; kb < K; kb += 4) {
        v2f a, b;
        a.x = arow[kb + khalf];
        a.y = arow[kb + khalf + 1];
        b.x = bmask * bcol[(size_t)(kb + khalf)     * Ncols];
        b.y = bmask * bcol[(size_t)(kb + khalf + 1) * Ncols];
        acc = __builtin_amdgcn_wmma_f32_16x16x4_f32(
            /*neg_a=*/false, a, /*neg_b=*/false, b,
            /*c_mod=*/(short)0, acc, /*reuse_a=*/false, /*reuse_b=*/false);
    }

    if (nvalid) {
        const float bv = bias[bn];
        const int mbase = m0 + ((lane >> 4) << 3);      // +8 for upper half lanes
#pragma unroll
        for (int v = 0; v < 8; ++v) {
            float val = acc[v] + bv;
            if (do_relu) val = fmaxf(val, 0.0f);
            C[(size_t)(mbase + v) * Ncols + bn] = val;
        }
    }
}

// ---------------- per-node: reciprocal L2 norm + init softmax stats ----------------
__global__ void node_prep(const float* __restrict__ h, float* __restrict__ rinv,
                          float* __restrict__ mmax, float* __restrict__ ssum, int Nn) {
    const int wave = (blockIdx.x * blockDim.x + threadIdx.x) >> 5;
    const int lane = threadIdx.x & 31;
    if (wave >= Nn) return;
    const float* row = h + (size_t)wave * HID;
    float a = row[lane];
    float b = row[lane + 32];
    float ss = waveReduceSum(a * a + b * b);
    if (lane == 0) {
        rinv[wave] = 1.0f / fmaxf(sqrtf(ss), EPS);
        mmax[wave] = -__builtin_inff();
        ssum[wave] = 0.0f;
    }
}

// ---------------- per-edge pass 1: cosine logit + segment max (atomic) ----------------
__global__ void edge_logits(const float* __restrict__ h, const float* __restrict__ rinv,
                            const int* __restrict__ ei, float* __restrict__ elog,
                            float* __restrict__ mmax, int E, int Nn) {
    const int wave = (blockIdx.x * blockDim.x + threadIdx.x) >> 5;
    const int lane = threadIdx.x & 31;
    const int Etot = E + Nn;
    if (wave >= Etot) return;
    int s, d;
    if (wave < E) { s = ei[wave]; d = ei[E + wave]; }
    else          { s = d = wave - E; }
    const float* hs = h + (size_t)s * HID;
    const float* hd = h + (size_t)d * HID;
    float p = hs[lane] * hd[lane] + hs[lane + 32] * hd[lane + 32];
    p = waveReduceSum(p);
    if (lane == 0) {
        float logit = p * rinv[s] * rinv[d];
        elog[wave] = logit;
        atomicMaxF(mmax + d, logit);
    }
}

// ---------------- per-edge pass 2: exp(logit - max) + segment sum ----------------
__global__ void edge_expsum(const int* __restrict__ ei, float* __restrict__ elog,
                            const float* __restrict__ mmax, float* __restrict__ ssum,
                            int E, int Nn) {
    const int e = blockIdx.x * blockDim.x + threadIdx.x;
    const int Etot = E + Nn;
    if (e >= Etot) return;
    const int d = (e < E) ? ei[E + e] : (e - E);
    const float v = __expf(elog[e] - mmax[d]);
    elog[e] = v;
    atomicAdd(ssum + d, v);
}

// ---------------- per-edge pass 3: scatter alpha * h[src] into h_out[dst] ----------------
__global__ void edge_aggr(const float* __restrict__ hin, float* __restrict__ hout,
                          const int* __restrict__ ei, const float* __restrict__ elog,
                          const float* __restrict__ ssum, int E, int Nn) {
    const int wave = (blockIdx.x * blockDim.x + threadIdx.x) >> 5;
    const int lane = threadIdx.x & 31;
    const int Etot = E + Nn;
    if (wave >= Etot) return;
    int s, d;
    if (wave < E) { s = ei[wave]; d = ei[E + wave]; }
    else          { s = d = wave - E; }
    const float alpha = elog[wave] / ssum[d];
    const float* hs = hin + (size_t)s * HID;
    float* ho = hout + (size_t)d * HID;
    atomicAdd(ho + lane,      alpha * hs[lane]);
    atomicAdd(ho + lane + 32, alpha * hs[lane + 32]);
}

// ---------------- per-row log_softmax (in place on d_out, C=40) ----------------
__global__ void row_log_softmax(float* __restrict__ out, int Nn) {
    const int wave = (blockIdx.x * blockDim.x + threadIdx.x) >> 5;
    const int lane = threadIdx.x & 31;
    if (wave >= Nn) return;
    float* row = out + (size_t)wave * NCLS;
    const float a = row[lane];                                     // lane < 32 < 40
    const float b = (lane < NCLS - 32) ? row[lane + 32] : -__builtin_inff();
    const float mx = waveReduceMax(fmaxf(a, b));
    float se = __expf(a - mx) + ((lane < NCLS - 32) ? __expf(b - mx) : 0.0f);
    se = waveReduceSum(se);
    const float lse = __logf(se);
    row[lane] = a - mx - lse;
    if (lane < NCLS - 32) row[lane + 32] = b - mx - lse;
}

// ================================================================================
extern "C" void kernel_launch(void* const* d_in, const int* in_sizes, int n_in,
                              void* d_out, int out_size, void* d_ws, size_t ws_size,
                              hipStream_t stream) {
    const float* x  = (const float*)d_in[0];
    const float* W1 = (const float*)d_in[1];
    const float* b1 = (const float*)d_in[2];
    const float* W2 = (const float*)d_in[3];
    const float* b2 = (const float*)d_in[4];
    const int*   ei = (const int*)d_in[5];

    const int Nn = in_sizes[0] / FIN;      // 100000
    const int E  = in_sizes[5] / 2;        // 1600000
    const int Etot = E + Nn;

    // workspace partition (floats)
    float* hA   = (float*)d_ws;
    float* hB   = hA   + (size_t)Nn * HID;
    float* rinv = hB   + (size_t)Nn * HID;
    float* mmax = rinv + Nn;
    float* ssum = mmax + Nn;
    float* elog = ssum + Nn;

    const int BLK = 256;                    // 8 waves/block
    const int wavesPerBlk = BLK / 32;

    // ---- lin1 + relu : hA = relu(x @ W1 + b1), WMMA f32 ----
    {
        const int ntiles = HID / 16;                       // 4
        const int waves  = (Nn / 16) * ntiles;             // 25000
        const int blocks = (waves + wavesPerBlk - 1) / wavesPerBlk;
        gemm_wmma_f32<<<blocks, BLK, 0, stream>>>(x, W1, b1, hA, Nn, FIN, HID, ntiles, 1);
    }

    // ---- 4 AGNN layers ----
    float* hin = hA;
    float* hout = hB;
    const int nodeBlocks = (Nn + wavesPerBlk - 1) / wavesPerBlk;
    const int edgeWaveBlocks = (Etot + wavesPerBlk - 1) / wavesPerBlk;
    const int edgeThreadBlocks = (Etot + BLK - 1) / BLK;
    for (int layer = 0; layer < 4; ++layer) {
        node_prep<<<nodeBlocks, BLK, 0, stream>>>(hin, rinv, mmax, ssum, Nn);
        edge_logits<<<edgeWaveBlocks, BLK, 0, stream>>>(hin, rinv, ei, elog, mmax, E, Nn);
        edge_expsum<<<edgeThreadBlocks, BLK, 0, stream>>>(ei, elog, mmax, ssum, E, Nn);
        hipMemsetAsync(hout, 0, (size_t)Nn * HID * sizeof(float), stream);
        edge_aggr<<<edgeWaveBlocks, BLK, 0, stream>>>(hin, hout, ei, elog, ssum, E, Nn);
        float* t = hin; hin = hout; hout = t;
    }

    // ---- lin2 : d_out = hin @ W2 + b2, WMMA f32 (N=40 padded to 48 w/ guards) ----
    {
        const int ntiles = (NCLS + 15) / 16;               // 3
        const int waves  = (Nn / 16) * ntiles;             // 18750
        const int blocks = (waves + wavesPerBlk - 1) / wavesPerBlk;
        gemm_wmma_f32<<<blocks, BLK, 0, stream>>>(hin, W2, b2, (float*)d_out, Nn, HID, NCLS, ntiles, 0);
    }

    // ---- log_softmax in place ----
    row_log_softmax<<<nodeBlocks, BLK, 0, stream>>>((float*)d_out, Nn);
}